// SimpleModel_13821204759105
// MI455X (gfx1250) — compile-verified
//
#include <hip/hip_runtime.h>
#include <math.h>

typedef float v2f __attribute__((ext_vector_type(2)));
typedef float v8f __attribute__((ext_vector_type(8)));

#define NPTS   1024
#define NCLS   10
#define K52    52
#define MPAD   16        // pad head-matrix cols to 16 (cols 10..15 = 0)
#define INV_SQRT52 0.13867504905630728f
#define INV_SQRT3  0.57735026918962576f

// ---------------------------------------------------------------------------
// Kernel 0: fold W3 (6x52, applied as /sqrt(52)) and Wout (10x6) into one
// M[52][16] (row-major, padded to 16 cols with zeros).
// out[b,n] = sum_k tp0[b,k] * M[k,n] + bout[n]
// ---------------------------------------------------------------------------
__global__ void build_M(const float* __restrict__ W3,
                        const float* __restrict__ Wout,
                        float* __restrict__ M) {
    int t = blockIdx.x * blockDim.x + threadIdx.x;
    if (t >= K52 * MPAD) return;
    int k = t / MPAD;
    int n = t % MPAD;
    float acc = 0.0f;
    if (n < NCLS) {
        #pragma unroll
        for (int j = 0; j < 6; ++j)
            acc += W3[j * K52 + k] * Wout[n * 6 + j];
        acc *= INV_SQRT52;
    }
    M[t] = acc;
}

// ---------------------------------------------------------------------------
// Kernel 1: per-batch reduction over 1024 points -> tp0[b][52].
// One 256-thread block (8 wave32) per batch; 4 points per thread in registers.
// ---------------------------------------------------------------------------
__global__ void __launch_bounds__(256)
reduce_feats(const float* __restrict__ pos,
             const float* __restrict__ W1s, const float* __restrict__ W1v,
             const float* __restrict__ W2s, const float* __restrict__ W2v,
             float* __restrict__ tp0) {
    const int b   = blockIdx.x;
    const int tid = threadIdx.x;               // 0..255
    const float* base = pos + (size_t)b * (NPTS * 3);

    // 12 consecutive floats per thread (4 points), coalesced 3x float4
    const float4* b4 = (const float4*)base;
    float4 q0 = b4[tid * 3 + 0];
    float4 q1 = b4[tid * 3 + 1];
    float4 q2 = b4[tid * 3 + 2];
    float px[4] = {q0.x, q0.w, q1.z, q2.y};
    float py[4] = {q0.y, q1.x, q1.w, q2.z};
    float pz[4] = {q0.z, q1.y, q2.x, q2.w};

    __shared__ float red[11][256];
    __shared__ float mean3[3];
    __shared__ float sv[11];

    // ---- pass 1: centroid ----
    float sx = 0.f, sy = 0.f, sz = 0.f;
    #pragma unroll
    for (int i = 0; i < 4; ++i) { sx += px[i]; sy += py[i]; sz += pz[i]; }
    red[0][tid] = sx; red[1][tid] = sy; red[2][tid] = sz;
    __syncthreads();
    for (int s = 128; s > 0; s >>= 1) {
        if (tid < s) {
            red[0][tid] += red[0][tid + s];
            red[1][tid] += red[1][tid + s];
            red[2][tid] += red[2][tid + s];
        }
        __syncthreads();
    }
    if (tid == 0) {
        mean3[0] = red[0][0] * (1.0f / NPTS);
        mean3[1] = red[1][0] * (1.0f / NPTS);
        mean3[2] = red[2][0] * (1.0f / NPTS);
    }
    __syncthreads();
    const float mx = mean3[0], my = mean3[1], mz = mean3[2];

    // ---- pass 2: rbf (8) + unit-direction (3) sums ----
    float acc[11];
    #pragma unroll
    for (int j = 0; j < 11; ++j) acc[j] = 0.f;
    #pragma unroll
    for (int i = 0; i < 4; ++i) {
        float dx = px[i] - mx, dy = py[i] - my, dz = pz[i] - mz;
        float r   = sqrtf(dx * dx + dy * dy + dz * dz);
        float inv = 1.0f / (r + 1e-12f);
        // centers = linspace(0,3.5,8) -> 0.5*j ; sigma = 0.5 ; exp(-2*(r-c)^2)
        #pragma unroll
        for (int j = 0; j < 8; ++j) {
            float d = r - 0.5f * (float)j;
            acc[j] += __expf(-2.0f * d * d);
        }
        acc[8]  += dx * inv;
        acc[9]  += dy * inv;
        acc[10] += dz * inv;
    }
    __syncthreads();   // done reading red[] from pass 1
    #pragma unroll
    for (int j = 0; j < 11; ++j) red[j][tid] = acc[j];
    __syncthreads();
    for (int s = 128; s > 0; s >>= 1) {
        if (tid < s) {
            #pragma unroll
            for (int j = 0; j < 11; ++j) red[j][tid] += red[j][tid + s];
        }
        __syncthreads();
    }
    if (tid < 11) sv[tid] = red[tid][0] * (1.0f / NPTS);
    __syncthreads();

    // ---- tiny per-batch tensor product -> tp0 row (52 values) ----
    if (tid < K52) {
        float val;
        if (tid < 36) {
            int a = tid / 6, c = tid % 6;
            float s1 = 0.f, s2 = 0.f;
            #pragma unroll
            for (int j = 0; j < 8; ++j) {
                s1 += sv[j] * W1s[a * 8 + j];
                s2 += sv[j] * W2s[c * 8 + j];
            }
            val = s1 * s2 * (1.0f / 8.0f);            // (1/sqrt8)*(1/sqrt8)
        } else {
            int j = tid - 36;
            int a = j / 4, c = j % 4;
            float vdot = sv[8] * sv[8] + sv[9] * sv[9] + sv[10] * sv[10];
            val = W1v[a] * W2v[c] * vdot * INV_SQRT3; // (v1.v2)/sqrt3
        }
        tp0[(size_t)b * K52 + tid] = val;
    }
}

// ---------------------------------------------------------------------------
// Kernel 2: out[8192,10] = tp0[8192,52] @ M[52,16] + bout, via
// V_WMMA_F32_16X16X4_F32 (13 K-steps of 4). One wave32 per 16-row tile.
// EXEC stays all-1s through all WMMA ops (masking only in the epilogue).
// ---------------------------------------------------------------------------
__global__ void __launch_bounds__(128)
wmma_head(const float* __restrict__ tp0, const float* __restrict__ M,
          const float* __restrict__ bout, float* __restrict__ out) {
    const int lane = threadIdx.x & 31;
    const int wave = threadIdx.x >> 5;
    const int row0 = (blockIdx.x * 4 + wave) * 16;

    const int mn = lane & 15;               // M index for A, N index for B/C/D
    const int kh = (lane >> 4) * 2;         // K half-select: 0 or 2

    v8f c = {0.f, 0.f, 0.f, 0.f, 0.f, 0.f, 0.f, 0.f};
    const float* arow = tp0 + (size_t)(row0 + mn) * K52;

    #pragma unroll
    for (int kb = 0; kb < 13; ++kb) {
        const int k0 = kb * 4 + kh;
        // A 16x4 f32 fragment: vgpr0 = A[m][k0], vgpr1 = A[m][k0+1]
        v2f a;  a.x  = arow[k0];
                a.y  = arow[k0 + 1];
        // B 4x16 f32 fragment: vgpr0 = B[k0][n], vgpr1 = B[k0+1][n]
        v2f bb; bb.x = M[k0 * MPAD + mn];
                bb.y = M[(k0 + 1) * MPAD + mn];
        // D = A x B + C   (neg_a, A, neg_b, B, c_mod, C, reuse_a, reuse_b)
        c = __builtin_amdgcn_wmma_f32_16x16x4_f32(
                false, a, false, bb, (short)0, c, false, false);
    }

    const float bias = (mn < NCLS) ? bout[mn] : 0.f;
    #pragma unroll
    for (int r = 0; r < 8; ++r) {
        const int mr = r + (lane >> 4) * 8;      // C/D layout: M = r (+8 hi half)
        if (mn < NCLS)
            out[(size_t)(row0 + mr) * NCLS + mn] = c[r] + bias;
    }
}

// ---------------------------------------------------------------------------
extern "C" void kernel_launch(void* const* d_in, const int* in_sizes, int n_in,
                              void* d_out, int out_size, void* d_ws, size_t ws_size,
                              hipStream_t stream) {
    const float* pos  = (const float*)d_in[0];
    const float* W1s  = (const float*)d_in[1];
    const float* W1v  = (const float*)d_in[2];
    const float* W2s  = (const float*)d_in[3];
    const float* W2v  = (const float*)d_in[4];
    const float* W3   = (const float*)d_in[5];
    const float* Wout = (const float*)d_in[6];
    const float* bout = (const float*)d_in[7];
    float* out = (float*)d_out;

    const int B = in_sizes[0] / (NPTS * 3);      // 8192

    float* ws  = (float*)d_ws;
    float* M   = ws;                              // 52*16 floats
    float* tp0 = ws + 1024;                       // B*52 floats

    build_M<<<dim3((K52 * MPAD + 255) / 256), dim3(256), 0, stream>>>(W3, Wout, M);
    reduce_feats<<<dim3(B), dim3(256), 0, stream>>>(pos, W1s, W1v, W2s, W2v, tp0);
    // one wave per 16 rows, 4 waves per block
    wmma_head<<<dim3(B / 64), dim3(128), 0, stream>>>(tp0, M, bout, out);
}